// TripletLoss_82755429860233
// MI455X (gfx1250) — compile-verified
//
#include <hip/hip_runtime.h>
#include <hip/hip_bf16.h>

// Triplet loss (batch-hard), fused:  E:[4096,128] f32, labels:[4096] int.
//   G = E E^T via V_WMMA_F32_16X16X4_F32 (fp32 WMMA: exact match to reference
//   numerics -- the d2 = |a|^2+|b|^2-2ab term cancels catastrophically and
//   feeds min/max selection, so low-precision matrix ops are not acceptable).
//   Row reductions (masked max / sum / cnt / min) fused into the GEMM epilogue;
//   the 64MB distance matrix is never materialized (only 2MB of HBM traffic).

typedef __attribute__((ext_vector_type(2))) float v2f;
typedef __attribute__((ext_vector_type(8))) float v8f;

#define N_EMB   4096
#define D_EMB   128
#define KSTEPS  (D_EMB / 4)   // 32 chained 16x16x4 WMMAs per tile
#define MARGIN_F 0.3f
#define EPS_F    1e-12f

// ---------------------------------------------------------------------------
// Kernel 1: squared norms, one wave32 per row (128 floats = 32 lanes x float4)
// ---------------------------------------------------------------------------
__global__ void tl_norms(const float* __restrict__ E, float* __restrict__ sq) {
    const int wave = (blockIdx.x * blockDim.x + threadIdx.x) >> 5;  // 0..4095
    const int lane = threadIdx.x & 31;
    const float4* row = (const float4*)(E + (size_t)wave * D_EMB);
    float4 v = row[lane];
    float s = v.x * v.x + v.y * v.y + v.z * v.z + v.w * v.w;
#pragma unroll
    for (int m = 16; m >= 1; m >>= 1) s += __shfl_xor(s, m, 32);
    if (lane == 0) sq[wave] = s;
}

// ---------------------------------------------------------------------------
// Kernel 2: fused WMMA-GEMM + masked row reductions.
// One wave per 16-row M-tile (256 waves total). Each wave sweeps all 256
// 16-column N-tiles; per tile: 32x V_WMMA_F32_16X16X4_F32 over K=128.
//
// Fragment layouts (ISA 7.12.2, wave32):
//   A 16x4 f32 : lanes 0-15 -> M=lane, {v0,v1}={K0,K1}; lanes 16-31 -> {K2,K3}
//   B  4x16 f32: lanes 0-15 -> N=lane, {v0,v1}={K0,K1}; lanes 16-31 -> {K2,K3}
//   C/D 16x16  : VGPR r -> row M=r (lanes 0-15) / M=r+8 (lanes 16-31), N=lane%16
// ---------------------------------------------------------------------------
__launch_bounds__(128, 1)
__global__ void tl_main(const float* __restrict__ E,
                        const int*   __restrict__ labels,
                        const float* __restrict__ sq,
                        float*       __restrict__ rowLoss) {
    const int lane = threadIdx.x & 31;
    const int half = lane >> 4;      // 0: low 16 lanes, 1: high 16 lanes
    const int l16  = lane & 15;
    const int wave = (blockIdx.x * blockDim.x + threadIdx.x) >> 5;  // 0..255
    const int i0   = wave << 4;      // first row of this wave's M-tile

    // Preload all A fragments for this M-tile once; reused for all 256 N-tiles.
    v2f a[KSTEPS];
    {
        const float* ar = E + (size_t)(i0 + l16) * D_EMB + 2 * half;
#pragma unroll
        for (int kk = 0; kk < KSTEPS; ++kk)
            a[kk] = *(const v2f*)(ar + 4 * kk);
    }

    // This lane's 8 rows: M = r + 8*half.
    int   lr[8];
    float sqa[8];
#pragma unroll
    for (int r = 0; r < 8; ++r) {
        lr[r]  = labels[i0 + r + 8 * half];
        sqa[r] = sq[i0 + r + 8 * half];
    }

    float posMax[8], posSum[8], posCnt[8], negMin[8];
#pragma unroll
    for (int r = 0; r < 8; ++r) {
        posMax[r] = -__builtin_inff();
        posSum[r] = 0.0f;
        posCnt[r] = 0.0f;
        negMin[r] = __builtin_inff();
    }

    for (int j0 = 0; j0 < N_EMB; j0 += 16) {
        const float* br = E + (size_t)(j0 + l16) * D_EMB + 2 * half;

        // CDNA5 prefetch of next tile's B rows (global_prefetch_b8).
        if (j0 + 16 < N_EMB)
            __builtin_prefetch(E + (size_t)(j0 + 16 + l16) * D_EMB, 0, 0);

        v8f c = {0.0f, 0.0f, 0.0f, 0.0f, 0.0f, 0.0f, 0.0f, 0.0f};
#pragma unroll
        for (int kk = 0; kk < KSTEPS; ++kk) {
            v2f b = *(const v2f*)(br + 4 * kk);
            // (neg_a, A, neg_b, B, c_mod, C, reuse_a, reuse_b)
            c = __builtin_amdgcn_wmma_f32_16x16x4_f32(
                    false, a[kk], false, b, (short)0, c, false, false);
        }

        const float sqb = sq[j0 + l16];
        const int   lb  = labels[j0 + l16];
#pragma unroll
        for (int r = 0; r < 8; ++r) {
            float d2   = sqa[r] + sqb - 2.0f * c[r];
            float dist = __builtin_sqrtf(fmaxf(d2, EPS_F));
            bool  same = (lr[r] == lb);
            posMax[r] = same ? fmaxf(posMax[r], dist) : posMax[r];
            posSum[r] = same ? (posSum[r] + dist)     : posSum[r];
            posCnt[r] = same ? (posCnt[r] + 1.0f)     : posCnt[r];
            negMin[r] = same ? negMin[r] : fminf(negMin[r], dist);
        }
    }

    // Reduce across the 16 lanes of each half (xor masks 1..8 stay in-half).
#pragma unroll
    for (int r = 0; r < 8; ++r) {
#pragma unroll
        for (int m = 8; m >= 1; m >>= 1) {
            posMax[r] = fmaxf(posMax[r], __shfl_xor(posMax[r], m, 32));
            posSum[r] += __shfl_xor(posSum[r], m, 32);
            posCnt[r] += __shfl_xor(posCnt[r], m, 32);
            negMin[r] = fminf(negMin[r], __shfl_xor(negMin[r], m, 32));
        }
    }

    if (l16 == 0) {  // lane 0 -> rows i0..i0+7, lane 16 -> rows i0+8..i0+15
#pragma unroll
        for (int r = 0; r < 8; ++r) {
            const int row = i0 + r + 8 * half;
            float pc  = posCnt[r];                    // >= 1 (diagonal)
            float nc  = (float)N_EMB - pc;
            float dap = (pc > 1.5f) ? posMax[r] : (posSum[r] / pc);
            float dan = (nc > 0.5f) ? negMin[r] : 0.0f;
            rowLoss[row] = fmaxf(dap - dan + MARGIN_F, 0.0f);
        }
    }
}

// ---------------------------------------------------------------------------
// Kernel 3: deterministic mean over 4096 per-row losses (no float atomics).
// ---------------------------------------------------------------------------
__global__ void tl_reduce(const float* __restrict__ rowLoss,
                          float* __restrict__ out) {
    __shared__ float s[256];
    float acc = 0.0f;
    for (int i = threadIdx.x; i < N_EMB; i += 256) acc += rowLoss[i];
    s[threadIdx.x] = acc;
    __syncthreads();
    for (int m = 128; m >= 1; m >>= 1) {
        if ((int)threadIdx.x < m) s[threadIdx.x] += s[threadIdx.x + m];
        __syncthreads();
    }
    if (threadIdx.x == 0) out[0] = s[0] * (1.0f / (float)N_EMB);
}

// ---------------------------------------------------------------------------
extern "C" void kernel_launch(void* const* d_in, const int* in_sizes, int n_in,
                              void* d_out, int out_size, void* d_ws, size_t ws_size,
                              hipStream_t stream) {
    (void)in_sizes; (void)n_in; (void)out_size; (void)ws_size;
    const float* E      = (const float*)d_in[0];
    const int*   labels = (const int*)d_in[1];
    float*       out    = (float*)d_out;

    float* sq      = (float*)d_ws;          // 4096 floats
    float* rowLoss = sq + N_EMB;            // 4096 floats

    // 4096 waves (one per row), 8 waves/block.
    tl_norms<<<512, 256, 0, stream>>>(E, sq);
    // 256 waves (one per 16-row tile), 4 waves/block.
    tl_main<<<64, 128, 0, stream>>>(E, labels, sq, rowLoss);
    // Final deterministic mean.
    tl_reduce<<<1, 256, 0, stream>>>(rowLoss, out);
}